// JetGNN_33792802685683
// MI455X (gfx1250) — compile-verified
//
#include <hip/hip_runtime.h>
#include <hip/hip_bf16.h>

typedef __attribute__((ext_vector_type(2))) float v2f;
typedef __attribute__((ext_vector_type(8))) float v8f;

#define N_NODES  100000
#define N_EDGES  1250000
#define N_GRAPHS 128

// ---------------------------------------------------------------- zero fill
__global__ void zero_f32(float* __restrict__ p, int n) {
    int i = blockIdx.x * blockDim.x + threadIdx.x;
    if (i < n) p[i] = 0.0f;
}

// ------------------------------------------------- layer 1: edge scatter-add
// agg1[dst] += x[src], x is [N,5]
__global__ void scatter1(const float* __restrict__ x,
                         const long long* __restrict__ ei,
                         float* __restrict__ agg1) {
    int e = blockIdx.x * blockDim.x + threadIdx.x;
    if (e >= N_EDGES) return;
    long long s = ei[e];
    long long d = ei[N_EDGES + e];
    const float* xs = x + s * 5;
    float* ad = agg1 + d * 5;
#pragma unroll
    for (int k = 0; k < 5; ++k) atomicAdd(ad + k, xs[k]);
}

// -------------------------------------- layer 1: node transform (K=5, VALU)
// h1 = relu(agg1 @ W1_rel + b1 + x @ W1_root)   [N,64]
__global__ void node1(const float* __restrict__ x,
                      const float* __restrict__ agg1,
                      const float* __restrict__ W1_rel,
                      const float* __restrict__ b1,
                      const float* __restrict__ W1_root,
                      float* __restrict__ h1) {
    int t = blockIdx.x * blockDim.x + threadIdx.x;
    if (t >= N_NODES * 64) return;
    int i = t >> 6;
    int j = t & 63;
    float acc = b1[j];
#pragma unroll
    for (int k = 0; k < 5; ++k) {
        acc = fmaf(agg1[i * 5 + k], W1_rel[k * 64 + j], acc);
        acc = fmaf(x[i * 5 + k],    W1_root[k * 64 + j], acc);
    }
    h1[t] = fmaxf(acc, 0.0f);
}

// ------------------------------------------------- layer 2: edge scatter-add
// agg2[dst] += h1[src], 64 feats; thread = (edge, 4-feature quad)
__global__ void scatter2(const float* __restrict__ h1,
                         const long long* __restrict__ ei,
                         float* __restrict__ agg2) {
    int t = blockIdx.x * blockDim.x + threadIdx.x;
    if (t >= N_EDGES * 16) return;
    int e  = t >> 4;
    int c0 = (t & 15) * 4;
    long long s = ei[e];
    long long d = ei[N_EDGES + e];
    const float4 v = *(const float4*)(h1 + s * 64 + c0);
    float* ad = agg2 + d * 64 + c0;
    atomicAdd(ad + 0, v.x);
    atomicAdd(ad + 1, v.y);
    atomicAdd(ad + 2, v.z);
    atomicAdd(ad + 3, v.w);
}

// ------------------------- layer 2: fused dual GEMM with f32 WMMA (16x16x4)
// h2 = relu(agg2 @ W2_rel + b2 + h1 @ W2_root)
// One wave owns a 16-row x 64-col output tile. N_NODES = 16*6250 exactly,
// so EXEC stays all-1s (WMMA requirement) with no tail handling.
//
// Weights are staged in LDS TRANSPOSED with a +4 float row pad:
//   sWT[col*68 + k] = W[k*64 + col]
// so each B fragment {W[k][col], W[k+1][col]} is one aligned ds_load_b64,
// and the per-lane bank index (4*col + k) mod 64 is conflict-free across
// the wave (lanes 0-15 use k, lanes 16-31 use k+2).
#define WPAD 68

__global__ __launch_bounds__(64) void layer2_wmma(
        const float* __restrict__ agg2, const float* __restrict__ h1,
        const float* __restrict__ W2_rel, const float* __restrict__ b2,
        const float* __restrict__ W2_root, float* __restrict__ h2) {
    __shared__ float sWrelT[64 * WPAD];
    __shared__ float sWrootT[64 * WPAD];
    int tid = threadIdx.x;
    for (int i = tid; i < 64 * 64; i += 64) {
        int k = i >> 6;          // source row    (K index)
        int c = i & 63;          // source column (N index)
        sWrelT[c * WPAD + k]  = W2_rel[i];
        sWrootT[c * WPAD + k] = W2_root[i];
    }
    __syncthreads();

    int wave = tid >> 5;
    int lane = tid & 31;
    int tile = blockIdx.x * 2 + wave;      // 3125 blocks * 2 waves = 6250 tiles
    int row0 = tile * 16;

    int m  = lane & 15;                    // A-matrix row within tile
    int kb = (lane >> 4) * 2;              // lanes 16-31 hold K+2,K+3

    v8f acc[4] = {};                       // 4 N-tiles of 16x16 f32

    const float* aggRow = agg2 + (long)(row0 + m) * 64;
    const float* hRow   = h1   + (long)(row0 + m) * 64;
    int colw = lane & 15;                  // B-matrix column within tile

    for (int ks = 0; ks < 16; ++ks) {
        int k = ks * 4 + kb;
        v2f aR = *(const v2f*)(aggRow + k);   // global_load_b64
        v2f aH = *(const v2f*)(hRow + k);     // global_load_b64
#pragma unroll
        for (int nt = 0; nt < 4; ++nt) {
            int col = nt * 16 + colw;
            v2f bR = *(const v2f*)&sWrelT[col * WPAD + k];   // ds_load_b64
            v2f bH = *(const v2f*)&sWrootT[col * WPAD + k];  // ds_load_b64
            acc[nt] = __builtin_amdgcn_wmma_f32_16x16x4_f32(
                false, aR, false, bR, (short)0, acc[nt], false, false);
            acc[nt] = __builtin_amdgcn_wmma_f32_16x16x4_f32(
                false, aH, false, bH, (short)0, acc[nt], false, false);
        }
    }

    // C/D layout: VGPR v, lanes 0-15 -> M=v, lanes 16-31 -> M=v+8; N=lane&15
    int rbase = row0 + (lane >> 4) * 8;
#pragma unroll
    for (int nt = 0; nt < 4; ++nt) {
        int col = nt * 16 + colw;
        float bias = b2[col];
#pragma unroll
        for (int v = 0; v < 8; ++v) {
            h2[(long)(rbase + v) * 64 + col] = fmaxf(acc[nt][v] + bias, 0.0f);
        }
    }
}

// ------------------------------------------------------ mean pool (atomics)
__global__ void pool(const float* __restrict__ h2,
                     const long long* __restrict__ batch,
                     float* __restrict__ sums, float* __restrict__ cnt) {
    int t = blockIdx.x * blockDim.x + threadIdx.x;
    if (t >= N_NODES * 16) return;
    int i  = t >> 4;
    int c0 = (t & 15) * 4;
    long long b = batch[i];
    const float4 v = *(const float4*)(h2 + (long)i * 64 + c0);
    float* sd = sums + b * 64 + c0;
    atomicAdd(sd + 0, v.x);
    atomicAdd(sd + 1, v.y);
    atomicAdd(sd + 2, v.z);
    atomicAdd(sd + 3, v.w);
    if ((t & 15) == 0) atomicAdd(cnt + b, 1.0f);
}

// ------------------------------------------------------------- final MLP
__global__ void mlp(const float* __restrict__ sums, const float* __restrict__ cnt,
                    const float* __restrict__ Wl1, const float* __restrict__ bl1,
                    const float* __restrict__ Wl2, const float* __restrict__ bl2,
                    float* __restrict__ out) {
    int g = blockIdx.x * blockDim.x + threadIdx.x;
    if (g >= N_GRAPHS) return;
    float inv = 1.0f / fmaxf(cnt[g], 1.0f);
    float gv[64];
#pragma unroll
    for (int k = 0; k < 64; ++k) gv[k] = sums[g * 64 + k] * inv;
    float hid[32];
#pragma unroll
    for (int j = 0; j < 32; ++j) {
        float a = bl1[j];
#pragma unroll
        for (int k = 0; k < 64; ++k) a = fmaf(gv[k], Wl1[k * 32 + j], a);
        hid[j] = fmaxf(a, 0.0f);
    }
#pragma unroll
    for (int o = 0; o < 2; ++o) {
        float a = bl2[o];
#pragma unroll
        for (int j = 0; j < 32; ++j) a = fmaf(hid[j], Wl2[j * 2 + o], a);
        out[g * 2 + o] = a;
    }
}

extern "C" void kernel_launch(void* const* d_in, const int* in_sizes, int n_in,
                              void* d_out, int out_size, void* d_ws, size_t ws_size,
                              hipStream_t stream) {
    const float*     x       = (const float*)d_in[0];
    const long long* ei      = (const long long*)d_in[1];
    const long long* batch   = (const long long*)d_in[2];
    const float*     W1_rel  = (const float*)d_in[3];
    const float*     b1      = (const float*)d_in[4];
    const float*     W1_root = (const float*)d_in[5];
    const float*     W2_rel  = (const float*)d_in[6];
    const float*     b2      = (const float*)d_in[7];
    const float*     W2_root = (const float*)d_in[8];
    const float*     Wl1     = (const float*)d_in[9];
    const float*     bl1     = (const float*)d_in[10];
    const float*     Wl2     = (const float*)d_in[11];
    const float*     bl2     = (const float*)d_in[12];
    float*           out     = (float*)d_out;

    // workspace layout (floats)
    float* ws   = (float*)d_ws;
    float* agg1 = ws;                       // 500,000
    float* h1   = agg1 + 500000;            // 6,400,000
    float* agg2 = h1   + 6400000;           // 6,400,000
    float* h2   = agg2 + 6400000;           // 6,400,000
    float* sums = h2   + 6400000;           // 8,192
    float* cnt  = sums + 8192;              // 128  (sums+cnt contiguous)

    const int B = 256;

    // zero the accumulation buffers (must happen every call)
    zero_f32<<<(500000 + B - 1) / B, B, 0, stream>>>(agg1, 500000);
    zero_f32<<<(6400000 + B - 1) / B, B, 0, stream>>>(agg2, 6400000);
    zero_f32<<<(8320 + B - 1) / B, B, 0, stream>>>(sums, 8320);

    // layer 1
    scatter1<<<(N_EDGES + B - 1) / B, B, 0, stream>>>(x, ei, agg1);
    node1<<<(N_NODES * 64 + B - 1) / B, B, 0, stream>>>(x, agg1, W1_rel, b1, W1_root, h1);

    // layer 2
    scatter2<<<(N_EDGES * 16 + B - 1) / B, B, 0, stream>>>(h1, ei, agg2);
    layer2_wmma<<<3125, 64, 0, stream>>>(agg2, h1, W2_rel, b2, W2_root, h2);

    // pool + head
    pool<<<(N_NODES * 16 + B - 1) / B, B, 0, stream>>>(h2, batch, sums, cnt);
    mlp<<<1, 128, 0, stream>>>(sums, cnt, Wl1, bl1, Wl2, bl2, out);
}